// SparseGCN_47132971106900
// MI455X (gfx1250) — compile-verified
//
#include <hip/hip_runtime.h>

typedef __attribute__((ext_vector_type(2))) float v2f;
typedef __attribute__((ext_vector_type(8))) float v8f;

#define NODES  50000
#define EDGES  800000
#define INDIM  128
#define FEAT   64      // HID == OUT == 64

// ---------------------------------------------------------------- utilities
__global__ void zero_f32(float* __restrict__ p, int n) {
  int i = blockIdx.x * blockDim.x + threadIdx.x;
  if (i < n) p[i] = 0.0f;
}

__global__ void init_deg(float* __restrict__ deg) {
  int i = blockIdx.x * blockDim.x + threadIdx.x;
  if (i < NODES) deg[i] = 1.0f;              // self-loop contribution
}

__global__ void accum_deg(const int* __restrict__ ei, float* __restrict__ deg) {
  int e = blockIdx.x * blockDim.x + threadIdx.x;
  if (e < EDGES) atomicAdd(&deg[ei[EDGES + e]], 1.0f);   // deg over dst column
}

__global__ void deg_to_dinv(float* __restrict__ deg) {
  int i = blockIdx.x * blockDim.x + threadIdx.x;
  if (i < NODES) {
    float d = deg[i];
    deg[i] = (d > 0.0f) ? rsqrtf(d) : 0.0f;
  }
}

// ---------------------------------------------------------------- WMMA GEMM
// C[NODES x 64] = A[NODES x K] * W[K x 64], full fp32 via V_WMMA_F32_16X16X4_F32.
// Block = 128 threads = 4 waves; wave w computes the 16x16 tile at
// rows [blockIdx.x*16, +16), cols [w*16, +16).  NODES % 16 == 0 -> no tails,
// EXEC is all-ones for every wave (WMMA requirement).
template <int K>
__global__ void __launch_bounds__(128)
gemm_wmma_f32(const float* __restrict__ A, const float* __restrict__ W,
              float* __restrict__ C) {
  const int lane = threadIdx.x & 31;
  const int wave = threadIdx.x >> 5;
  const int m0   = blockIdx.x * 16;
  const int n0   = wave * 16;
  const int half = lane >> 4;     // 0: lanes 0-15, 1: lanes 16-31
  const int l15  = lane & 15;
  const int koff = half * 2;      // upper half-wave holds K+2, K+3

  const float* __restrict__ arow = A + (size_t)(m0 + l15) * K + koff;
  const int col = n0 + l15;

  v8f acc = {};
#pragma unroll 4
  for (int k = 0; k < K; k += 4) {
    v2f a, b;
    a.x = arow[k];
    a.y = arow[k + 1];
    b.x = W[(k + koff) * FEAT + col];
    b.y = W[(k + koff + 1) * FEAT + col];
    // D = A(16x4) * B(4x16) + C, fp32 in / fp32 accumulate
    acc = __builtin_amdgcn_wmma_f32_16x16x4_f32(
        /*neg_a=*/false, a, /*neg_b=*/false, b,
        /*c_mod=*/(short)0, acc, /*reuse_a=*/false, /*reuse_b=*/false);
  }

  const int orow = m0 + half * 8;   // D VGPR r -> row m0+r (lanes 0-15), m0+8+r (16-31)
#pragma unroll
  for (int r = 0; r < 8; ++r)
    C[(size_t)(orow + r) * FEAT + col] = acc[r];
}

// ---------------------------------------------------------------- scatter-add
// One edge per 64 threads (one thread per feature): coalesced read of the
// source row, atomic f32 add into the destination row (L2-resident).
__global__ void __launch_bounds__(256)
scatter_edges(const int* __restrict__ ei, const float* __restrict__ dinv,
              const float* __restrict__ h, float* __restrict__ out) {
  int t = blockIdx.x * 256 + threadIdx.x;
  int e = t >> 6;
  int f = t & 63;
  if (e >= EDGES) return;
  int s = ei[e];
  int d = ei[EDGES + e];
  float w = dinv[s] * dinv[d];
  atomicAdd(&out[(size_t)d * FEAT + f], w * h[(size_t)s * FEAT + f]);
}

// ---------------------------------------------------------------- epilogue
// out = (RELU?) ( agg + dinv[i]^2 * h + bias )   -- folds the self-loop edge
// (norm = dinv[i]*dinv[i]) so we never pay atomics for the N loop edges.
template <bool RELU>
__global__ void __launch_bounds__(256)
finalize_layer(const float* __restrict__ agg, const float* __restrict__ h,
               const float* __restrict__ dinv, const float* __restrict__ bias,
               float* __restrict__ out) {
  int idx = blockIdx.x * 256 + threadIdx.x;
  if (idx >= NODES * FEAT) return;
  int i = idx >> 6;
  int f = idx & 63;
  float di = dinv[i];
  float v  = agg[idx] + di * di * h[idx] + bias[f];
  out[idx] = RELU ? fmaxf(v, 0.0f) : v;
}

// ---------------------------------------------------------------- launcher
extern "C" void kernel_launch(void* const* d_in, const int* in_sizes, int n_in,
                              void* d_out, int out_size, void* d_ws, size_t ws_size,
                              hipStream_t stream) {
  const float* x  = (const float*)d_in[0];
  const int*   ei = (const int*)d_in[1];   // [2, E] row-major, int32
  const float* W1 = (const float*)d_in[2];
  const float* b1 = (const float*)d_in[3];
  const float* W2 = (const float*)d_in[4];
  const float* b2 = (const float*)d_in[5];
  const float* W3 = (const float*)d_in[6];
  const float* b3 = (const float*)d_in[7];
  float* out = (float*)d_out;

  float* ws   = (float*)d_ws;
  float* dinv = ws;                                 // [NODES]
  float* P    = ws + NODES;                         // [NODES*64] pre-agg h
  float* Q    = P + (size_t)NODES * FEAT;           // [NODES*64] agg / activations

  const int nThreads  = 256;
  const int gNodes    = (NODES + nThreads - 1) / nThreads;
  const int gEdges    = (EDGES + nThreads - 1) / nThreads;
  const int gFeat     = (NODES * FEAT) / nThreads;  // 12500, exact
  const int gScatter  = (EDGES * FEAT) / nThreads;  // 200000, exact
  const int gGemm     = NODES / 16;                 // 3125, exact

  // --- degree / normalization ---
  init_deg<<<gNodes, nThreads, 0, stream>>>(dinv);
  accum_deg<<<gEdges, nThreads, 0, stream>>>(ei, dinv);
  deg_to_dinv<<<gNodes, nThreads, 0, stream>>>(dinv);

  // --- layer 1: x @ W1 -> P ; aggregate -> Q ; +self-loop +b1, ReLU ---
  gemm_wmma_f32<INDIM><<<gGemm, 128, 0, stream>>>(x, W1, P);
  zero_f32<<<gFeat, nThreads, 0, stream>>>(Q, NODES * FEAT);
  scatter_edges<<<gScatter, nThreads, 0, stream>>>(ei, dinv, P, Q);
  finalize_layer<true><<<gFeat, nThreads, 0, stream>>>(Q, P, dinv, b1, Q);

  // --- layer 2 ---
  gemm_wmma_f32<FEAT><<<gGemm, 128, 0, stream>>>(Q, W2, P);
  zero_f32<<<gFeat, nThreads, 0, stream>>>(Q, NODES * FEAT);
  scatter_edges<<<gScatter, nThreads, 0, stream>>>(ei, dinv, P, Q);
  finalize_layer<true><<<gFeat, nThreads, 0, stream>>>(Q, P, dinv, b2, Q);

  // --- layer 3 (no ReLU) -> d_out ---
  gemm_wmma_f32<FEAT><<<gGemm, 128, 0, stream>>>(Q, W3, P);
  zero_f32<<<gFeat, nThreads, 0, stream>>>(out, NODES * FEAT);
  scatter_edges<<<gScatter, nThreads, 0, stream>>>(ei, dinv, P, out);
  finalize_layer<false><<<gFeat, nThreads, 0, stream>>>(out, P, dinv, b3, out);
}